// OrthogonalMatchingPursuitSecondVersion_40432822125177
// MI455X (gfx1250) — compile-verified
//
#include <hip/hip_runtime.h>
#include <hip/hip_bf16.h>
#include <math.h>

typedef __attribute__((ext_vector_type(2))) float v2f;
typedef __attribute__((ext_vector_type(8))) float v8f;
typedef __attribute__((ext_vector_type(4))) unsigned int v4u;
typedef __attribute__((ext_vector_type(8))) int v8i;
typedef __attribute__((ext_vector_type(4))) int v4i;

#define BB   128
#define LL   512
#define NN   2048
#define NTOT 2049
#define KK   24
#define PN   2064   // padded proj row stride
#define NT   129    // ceil(2049/16) column tiles

__device__ __forceinline__ float softplus_f(float x) { return log1pf(expf(x)); }

// ---------------- init ----------------
__global__ __launch_bounds__(256) void init_kernel(const float* __restrict__ y,
                                                   float* __restrict__ R,
                                                   int* __restrict__ idx,
                                                   int* __restrict__ done) {
    int g = blockIdx.x * 256 + threadIdx.x;           // 0..65535
    R[g] = y[g];                                      // residuals = yv
    if (g < BB * KK) idx[g] = 0;                      // indices init zeros
    if (g == 0) done[0] = 0;
}

__global__ __launch_bounds__(256) void normy_kernel(const float* __restrict__ y,
                                                    float* __restrict__ nrmy) {
    __shared__ float red[256];
    int b = blockIdx.x, t = threadIdx.x;
    float s = 0.f;
    for (int l = t; l < LL; l += 256) { float v = y[(size_t)b * LL + l]; s += v * v; }
    red[t] = s; __syncthreads();
    for (int st = 128; st > 0; st >>= 1) { if (t < st) red[t] += red[t + st]; __syncthreads(); }
    if (t == 0) nrmy[b] = sqrtf(red[0]);
}

__global__ void normy_reduce(const float* __restrict__ nrmy, float* __restrict__ norm_y) {
    __shared__ float red[128];
    int t = threadIdx.x;
    red[t] = nrmy[t]; __syncthreads();
    for (int s = 64; s > 0; s >>= 1) { if (t < s) red[t] += red[t + s]; __syncthreads(); }
    if (t == 0) norm_y[0] = red[0] / (float)BB;
}

// ---------------- projection: P[b,n] = |sum_l R[b,l] * D[l,n]| via f32 WMMA ----------------
// Block = 256 threads (8 waves). Each block handles one 16-column tile of D for all 128 batches.
// D tile (512x16 f32, 32 KB) staged into LDS by the Tensor Data Mover (TDM), shared by 8 waves.
__global__ __launch_bounds__(256) void proj_kernel(const float* __restrict__ X,
                                                   const float* __restrict__ R,
                                                   float* __restrict__ P) {
    __shared__ float Dt[LL * 16];                     // 32 KB
    const int n0 = blockIdx.x * 16;
    const int t = threadIdx.x;

    // --- TDM: DMA the 512 x 16 tile D[:, n0:n0+16] (D = Xb[0]) into LDS ---
    if ((t >> 5) == 0) {                              // wave 0 issues the tensor load
        unsigned long long gaddr = (unsigned long long)(uintptr_t)(X + n0);
        unsigned int lds_base = (unsigned int)(uintptr_t)(&Dt[0]);   // addr[31:0] = LDS offset
        int remain = NN - n0; if (remain < 0) remain = 0;            // OOB columns read as 0

        v4u g0;
        g0[0] = 1u;                                                  // count=1, user descriptor
        g0[1] = lds_base;                                            // lds_addr
        g0[2] = (unsigned int)(gaddr & 0xFFFFFFFFu);                 // global_addr[31:0]
        g0[3] = (unsigned int)((gaddr >> 32) & 0x01FFFFFFu) | (2u << 30); // addr[56:32] | type=2

        v8i g1;
        g1[0] = (int)(2u << 16);                                     // data_size=2 (4 bytes)
        g1[1] = (int)(((unsigned)remain & 0xFFFFu) << 16);           // tensor_dim0[15:0]
        g1[2] = (int)((((unsigned)remain >> 16) & 0xFFFFu) |
                      (((unsigned)LL & 0xFFFFu) << 16));             // td0[31:16] | tensor_dim1[15:0]
        g1[3] = (int)(16u << 16);                                    // td1[31:16]=0 | tile_dim0=16
        g1[4] = (int)((unsigned)LL & 0xFFFFu);                       // tile_dim1=512, tile_dim2=0
        g1[5] = (int)NN;                                             // tensor_dim0_stride = 2048
        g1[6] = 0;                                                   // stride0[47:32], stride1[15:0]
        g1[7] = 0;                                                   // stride1[47:16]

        v4i z4 = {0, 0, 0, 0};                                       // 2-D tensor: groups 2/3 unused
        v8i z8 = {0, 0, 0, 0, 0, 0, 0, 0};
        __builtin_amdgcn_tensor_load_to_lds(g0, g1, z4, z4, z8, 0);
        __builtin_amdgcn_s_wait_tensorcnt(0);
    }
    __syncthreads();
    // patch appended ones column (n == NN) if it falls in this tile
    {
        int j1 = NN - n0;
        if (j1 >= 0 && j1 < 16)
            for (int l = t; l < LL; l += 256) Dt[l * 16 + j1] = 1.0f;
    }
    __syncthreads();

    const int lane = t & 31;
    const int wave = t >> 5;
    const int b0   = wave * 16;
    const int mrow = lane & 15;
    const int koff = (lane >> 4) << 1;                // 0 (lanes 0-15) or 2 (lanes 16-31)
    const float* __restrict__ Rrow = R + (size_t)(b0 + mrow) * LL;

    v8f acc = {};                                     // C = 0
#pragma unroll 4
    for (int k = 0; k < LL; k += 4) {
        // A fragment (16x4 f32): VGPR0 = A[M=mrow][K=k+koff], VGPR1 = K=k+koff+1
        v2f a = *(const v2f*)(Rrow + k + koff);
        // B fragment (4x16 f32): VGPR0 = B[K=k+koff][N=mrow], VGPR1 = K=k+koff+1
        v2f bm;
        bm.x = Dt[(k + koff) * 16 + mrow];
        bm.y = Dt[(k + koff + 1) * 16 + mrow];
        acc = __builtin_amdgcn_wmma_f32_16x16x4_f32(false, a, false, bm,
                                                    (short)0, acc, false, false);
    }

    const int ncol = n0 + mrow;
    if (ncol < NTOT) {
        const int half = lane >> 4;
#pragma unroll
        for (int r = 0; r < 8; ++r) {                 // C/D: VGPR r -> M = r (+8 for upper half)
            int brow = b0 + r + half * 8;
            P[(size_t)brow * PN + ncol] = fabsf(acc[r]);
        }
    }
}

// ---------------- per-batch argmax over NTOT ----------------
__global__ __launch_bounds__(256) void argmax_kernel(const float* __restrict__ P,
                                                     int* __restrict__ idx_cand) {
    __shared__ float sv[256]; __shared__ int si[256];
    int b = blockIdx.x, t = threadIdx.x;
    float best = -1.f; int bi = NTOT;
    const float* p = P + (size_t)b * PN;
    for (int n = t; n < NTOT; n += 256) {
        float v = p[n];
        if (v > best || (v == best && n < bi)) { best = v; bi = n; }
    }
    sv[t] = best; si[t] = bi; __syncthreads();
    for (int s = 128; s > 0; s >>= 1) {
        if (t < s) {
            float ov = sv[t + s]; int oi = si[t + s];
            if (ov > sv[t] || (ov == sv[t] && oi < si[t])) { sv[t] = ov; si[t] = oi; }
        }
        __syncthreads();
    }
    if (t == 0) idx_cand[b] = si[0];
}

// ---------------- per-batch Gram + regularized solve + residual ----------------
__global__ __launch_bounds__(256) void solve_kernel(const float* __restrict__ X,
                                                    const float* __restrict__ y,
                                                    const float* __restrict__ lam,
                                                    const int* __restrict__ idx,
                                                    const int* __restrict__ idx_cand,
                                                    float* __restrict__ Rn,
                                                    float* __restrict__ nrm,
                                                    int iter) {
    __shared__ float cols[KK * LL];                   // 48 KB: selected dictionary columns
    __shared__ float sy[LL];
    __shared__ float G[KK * KK];
    __shared__ float rhs[KK];
    __shared__ float sol[KK];
    __shared__ float red[256];
    const int b = blockIdx.x, t = threadIdx.x;
    const int m = iter + 1;

    for (int l = t; l < LL; l += 256) sy[l] = y[(size_t)b * LL + l];
    for (int e = t; e < m * LL; e += 256) {
        int k = e >> 9, l = e & 511;
        int c = (k < iter) ? idx[b * KK + k] : idx_cand[b];
        cols[e] = (c < NN) ? X[(size_t)l * NN + c] : 1.0f;   // atoms from D = Xb[0]
    }
    __syncthreads();

    const float reg = softplus_f(lam[0]);
    for (int p = t; p < m * m; p += 256) {
        int j = p / m, kq = p % m;
        if (j <= kq) {
            const float* cj = cols + j * LL; const float* ck = cols + kq * LL;
            float s = 0.f;
            for (int l = 0; l < LL; ++l) s += cj[l] * ck[l];
            if (j == kq) s += reg;
            G[j * m + kq] = s; G[kq * m + j] = s;
        }
    }
    for (int k = t; k < m; k += 256) {
        const float* ck = cols + k * LL;
        float s = 0.f;
        for (int l = 0; l < LL; ++l) s += ck[l] * sy[l];
        rhs[k] = s;
    }
    __syncthreads();

    // Gaussian elimination (SPD: gram + reg*I), rows in parallel
    for (int p = 0; p < m - 1; ++p) {
        if (t > p && t < m) {
            float f = G[t * m + p] / G[p * m + p];
            for (int c = p; c < m; ++c) G[t * m + c] -= f * G[p * m + c];
            rhs[t] -= f * rhs[p];
        }
        __syncthreads();
    }
    if (t == 0) {
        for (int p = m - 1; p >= 0; --p) {
            float s = rhs[p];
            for (int c = p + 1; c < m; ++c) s -= G[p * m + c] * sol[c];
            sol[p] = s / G[p * m + p];
        }
    }
    __syncthreads();

    float sq = 0.f;
    for (int l = t; l < LL; l += 256) {
        float r = sy[l];
        for (int k = 0; k < m; ++k) r -= cols[k * LL + l] * sol[k];
        Rn[(size_t)b * LL + l] = r;
        sq += r * r;
    }
    red[t] = sq; __syncthreads();
    for (int s = 128; s > 0; s >>= 1) { if (t < s) red[t] += red[t + s]; __syncthreads(); }
    if (t == 0) nrm[b] = sqrtf(red[0]);
}

// ---------------- early-stop bookkeeping (faithful freeze semantics) ----------------
__global__ void commit_kernel(const float* __restrict__ nrm, const float* __restrict__ norm_y,
                              int* __restrict__ done, int* __restrict__ commit) {
    __shared__ float red[128];
    int t = threadIdx.x;
    red[t] = nrm[t]; __syncthreads();
    for (int s = 64; s > 0; s >>= 1) { if (t < s) red[t] += red[t + s]; __syncthreads(); }
    if (t == 0) {
        commit[0] = (done[0] == 0) ? 1 : 0;           // commit uses OLD done
        float meanres = red[0] / (float)BB;
        if (meanres <= 0.001f * norm_y[0]) done[0] = 1;
    }
}

__global__ __launch_bounds__(256) void apply_kernel(const float* __restrict__ Rn,
                                                    float* __restrict__ R,
                                                    const int* __restrict__ idx_cand,
                                                    int* __restrict__ idx,
                                                    const int* __restrict__ commit,
                                                    int iter) {
    if (commit[0] == 0) return;
    int g = blockIdx.x * 256 + threadIdx.x;           // 0..65535
    R[g] = Rn[g];
    if (g < BB) idx[g * KK + iter] = idx_cand[g];
}

// ---------------- final refit vs per-batch Xb + duplicate-scatter masking ----------------
__global__ __launch_bounds__(256) void final_solve_kernel(const float* __restrict__ X,
                                                          const float* __restrict__ y,
                                                          const float* __restrict__ lam,
                                                          const int* __restrict__ idx,
                                                          float* __restrict__ eff) {
    __shared__ float cols[KK * LL];
    __shared__ float sy[LL];
    __shared__ float G[KK * KK];
    __shared__ float rhs[KK];
    __shared__ float sol[KK];
    const int b = blockIdx.x, t = threadIdx.x;
    const float* __restrict__ Xb = X + (size_t)b * LL * NN;

    for (int l = t; l < LL; l += 256) sy[l] = y[(size_t)b * LL + l];
    for (int e = t; e < KK * LL; e += 256) {
        int k = e >> 9, l = e & 511;
        int c = idx[b * KK + k];
        cols[e] = (c < NN) ? Xb[(size_t)l * NN + c] : 1.0f;  // atoms from per-batch Xb
    }
    __syncthreads();

    const float reg = softplus_f(lam[0]);
    for (int p = t; p < KK * KK; p += 256) {
        int j = p / KK, kq = p % KK;
        if (j <= kq) {
            const float* cj = cols + j * LL; const float* ck = cols + kq * LL;
            float s = 0.f;
            for (int l = 0; l < LL; ++l) s += cj[l] * ck[l];
            if (j == kq) s += reg;
            G[j * KK + kq] = s; G[kq * KK + j] = s;
        }
    }
    for (int k = t; k < KK; k += 256) {
        const float* ck = cols + k * LL;
        float s = 0.f;
        for (int l = 0; l < LL; ++l) s += ck[l] * sy[l];
        rhs[k] = s;
    }
    __syncthreads();

    for (int p = 0; p < KK - 1; ++p) {
        if (t > p && t < KK) {
            float f = G[t * KK + p] / G[p * KK + p];
            for (int c = p; c < KK; ++c) G[t * KK + c] -= f * G[p * KK + c];
            rhs[t] -= f * rhs[p];
        }
        __syncthreads();
    }
    if (t == 0) {
        for (int p = KK - 1; p >= 0; --p) {
            float s = rhs[p];
            for (int c = p + 1; c < KK; ++c) s -= G[p * KK + c] * sol[c];
            sol[p] = s / G[p * KK + p];
        }
        // scatter W.at[idx].set(coefs): last occurrence of a duplicate index wins
        for (int k = 0; k < KK; ++k) {
            int c = idx[b * KK + k];
            bool last = true;
            for (int k2 = k + 1; k2 < KK; ++k2)
                if (idx[b * KK + k2] == c) { last = false; break; }
            eff[b * KK + k] = last ? sol[k] : 0.f;
        }
    }
}

__global__ __launch_bounds__(256) void output_kernel(const float* __restrict__ X,
                                                     const int* __restrict__ idx,
                                                     const float* __restrict__ eff,
                                                     float* __restrict__ out) {
    __shared__ float se[KK]; __shared__ int si[KK];
    const int b = blockIdx.x, t = threadIdx.x;
    if (t < KK) { se[t] = eff[b * KK + t]; si[t] = idx[b * KK + t]; }
    __syncthreads();
    const float* __restrict__ Xb = X + (size_t)b * LL * NN;
    for (int l = t; l < LL; l += 256) {
        float s = 0.f;
#pragma unroll
        for (int k = 0; k < KK; ++k) {
            int c = si[k];
            float v = (c < NN) ? Xb[(size_t)l * NN + c] : 1.0f;
            s += se[k] * v;
        }
        out[(size_t)b * LL + l] = s;
    }
}

extern "C" void kernel_launch(void* const* d_in, const int* in_sizes, int n_in,
                              void* d_out, int out_size, void* d_ws, size_t ws_size,
                              hipStream_t stream) {
    (void)in_sizes; (void)n_in; (void)out_size; (void)ws_size;
    const float* X   = (const float*)d_in[0];   // (128,512,2048)
    const float* y   = (const float*)d_in[1];   // (128,512,1)
    const float* lam = (const float*)d_in[2];   // scalar

    char* base = (char*)d_ws;
    size_t off = 0;
    auto alloc = [&](size_t bytes) { void* p = base + off; off = (off + bytes + 255) & ~(size_t)255; return p; };
    float* R        = (float*)alloc(BB * LL * sizeof(float));
    float* Rn       = (float*)alloc(BB * LL * sizeof(float));
    float* P        = (float*)alloc((size_t)BB * PN * sizeof(float));
    int*   idx      = (int*)  alloc(BB * KK * sizeof(int));
    int*   idx_cand = (int*)  alloc(BB * sizeof(int));
    float* nrm      = (float*)alloc(BB * sizeof(float));
    float* nrmy     = (float*)alloc(BB * sizeof(float));
    float* norm_y   = (float*)alloc(sizeof(float));
    int*   done     = (int*)  alloc(sizeof(int));
    int*   commit   = (int*)  alloc(sizeof(int));
    float* eff      = (float*)alloc(BB * KK * sizeof(float));

    init_kernel<<<256, 256, 0, stream>>>(y, R, idx, done);
    normy_kernel<<<BB, 256, 0, stream>>>(y, nrmy);
    normy_reduce<<<1, 128, 0, stream>>>(nrmy, norm_y);

    for (int i = 0; i < KK; ++i) {
        proj_kernel<<<NT, 256, 0, stream>>>(X, R, P);
        argmax_kernel<<<BB, 256, 0, stream>>>(P, idx_cand);
        solve_kernel<<<BB, 256, 0, stream>>>(X, y, lam, idx, idx_cand, Rn, nrm, i);
        commit_kernel<<<1, 128, 0, stream>>>(nrm, norm_y, done, commit);
        apply_kernel<<<256, 256, 0, stream>>>(Rn, R, idx_cand, idx, commit, i);
    }

    final_solve_kernel<<<BB, 256, 0, stream>>>(X, y, lam, idx, eff);
    output_kernel<<<BB, 256, 0, stream>>>(X, idx, eff, (float*)d_out);
}